// NequIP_TimeEmbed_39024072851885
// MI455X (gfx1250) — compile-verified
//
#include <hip/hip_runtime.h>
#include <math.h>

// ---------------------------------------------------------------------------
// CDNA5 (gfx1250) implementation of the NequIP-style equivariant GNN forward.
// All GEMM-shaped contractions (radial MLP, fctp_scalar sc/lin1/lin2) run on
// V_WMMA_F32_16X16X4_F32 (fp32 WMMA, wave32, one wave per 16x16 C tile).
// Edge tiles handled by index clamping (no exec-masked loads in the K loop).
// ---------------------------------------------------------------------------

typedef __attribute__((ext_vector_type(2))) float v2f;
typedef __attribute__((ext_vector_type(8))) float v8f;

__device__ __forceinline__ float silu_f(float x)    { return x / (1.0f + __expf(-x)); }
__device__ __forceinline__ float sigmoid_f(float x) { return 1.0f / (1.0f + __expf(-x)); }
__device__ __forceinline__ float softplus_f(float x){ return (x > 30.0f) ? x : log1pf(__expf(x)); }
__device__ __forceinline__ int   clampi(int v, int hi) { return v < hi ? v : hi; }

// ------------------------- Clebsch-Gordan constants -------------------------
template<int NV> struct CGA { float v[NV]; };

constexpr float R3  = 0.57735027f;   // 1/sqrt(3)
constexpr float R5  = 0.44721360f;   // 1/sqrt(5)
constexpr float S6  = 0.40824829f;   // 1/sqrt(6)
constexpr float CA  = 0.31622777f;   // 1/sqrt(10)
constexpr float CB  = 0.18257419f;   // 1/sqrt(30)
constexpr float CTB = 0.36514837f;   // 2/sqrt(30)
constexpr float GG  = 0.18257419f;   // 1/sqrt(30) generator scale
constexpr float G3  = 0.31622777f;   // sqrt(3)/sqrt(30)
constexpr float HH  = 0.25819889f;   // 1/sqrt(15)

constexpr CGA<1>  mk000(){ CGA<1> a{}; a.v[0]=1.0f; return a; }
constexpr CGA<9>  mk011(){ CGA<9> a{}; for(int k=0;k<3;k++) a.v[k*3+k]=R3; return a; }
constexpr CGA<25> mk022(){ CGA<25> a{}; for(int k=0;k<5;k++) a.v[k*5+k]=R5; return a; }
constexpr CGA<9>  mk101(){ CGA<9> a{}; for(int k=0;k<3;k++) a.v[k*3+k]=R3; return a; }
constexpr CGA<9>  mk110(){ CGA<9> a{}; for(int i=0;i<3;i++) a.v[i*3+i]=R3; return a; }
constexpr CGA<25> mk202(){ CGA<25> a{}; for(int k=0;k<5;k++) a.v[k*5+k]=R5; return a; }
constexpr CGA<25> mk220(){ CGA<25> a{}; for(int i=0;i<5;i++) a.v[i*5+i]=R5; return a; }
constexpr CGA<27> mk111(){ CGA<27> a{};   // Levi-Civita / sqrt(6), idx (k*3+i)*3+j
  a.v[(0*3+1)*3+2]= S6; a.v[(0*3+2)*3+1]=-S6;
  a.v[(1*3+2)*3+0]= S6; a.v[(1*3+0)*3+2]=-S6;
  a.v[(2*3+0)*3+1]= S6; a.v[(2*3+1)*3+0]=-S6; return a; }
// (1,1,2): symmetric traceless; l1 order (y,z,x), l2=(y,z,x), l3=(xy,yz,z2,xz,x2-y2)
constexpr CGA<45> mk112(){ CGA<45> a{};
  a.v[(0*3+2)*3+0]= CA; a.v[(0*3+0)*3+2]= CA;
  a.v[(1*3+0)*3+1]= CA; a.v[(1*3+1)*3+0]= CA;
  a.v[(2*3+0)*3+0]=-CB; a.v[(2*3+1)*3+1]= CTB; a.v[(2*3+2)*3+2]=-CB;
  a.v[(3*3+1)*3+2]= CA; a.v[(3*3+2)*3+1]= CA;
  a.v[(4*3+0)*3+0]=-CA; a.v[(4*3+2)*3+2]= CA; return a; }
// (2,1,1): transpose of (1,1,2); idx (k*5+K)*3+j
constexpr CGA<45> mk211(){ CGA<45> a{};
  a.v[(0*5+0)*3+2]= CA; a.v[(0*5+1)*3+1]= CA; a.v[(0*5+2)*3+0]=-CB; a.v[(0*5+4)*3+0]=-CA;
  a.v[(1*5+1)*3+0]= CA; a.v[(1*5+2)*3+1]= CTB; a.v[(1*5+3)*3+2]= CA;
  a.v[(2*5+0)*3+0]= CA; a.v[(2*5+2)*3+2]=-CB; a.v[(2*5+3)*3+1]= CA; a.v[(2*5+4)*3+2]= CA;
  return a; }
// (2,1,2): angular-momentum-generator coupling; idx (k*5+K)*3+j, j:0=Ly,1=Lz,2=Lx
constexpr CGA<75> mk212(){ CGA<75> a{};
  a.v[(0*5+1)*3+0]=-GG; a.v[(1*5+0)*3+0]= GG; a.v[(2*5+3)*3+0]= G3;
  a.v[(3*5+2)*3+0]=-G3; a.v[(3*5+4)*3+0]=-GG; a.v[(4*5+3)*3+0]= GG;
  a.v[(0*5+4)*3+1]= 2*GG; a.v[(1*5+3)*3+1]=-GG; a.v[(3*5+1)*3+1]= GG; a.v[(4*5+0)*3+1]=-2*GG;
  a.v[(0*5+3)*3+2]= GG; a.v[(1*5+2)*3+2]= G3; a.v[(1*5+4)*3+2]=-GG;
  a.v[(2*5+1)*3+2]=-G3; a.v[(3*5+0)*3+2]=-GG; a.v[(4*5+1)*3+2]= GG; return a; }
// (2,2,1): generators again; idx (k*5+K)*5+Kp, k:0=Ly,1=Lz,2=Lx
constexpr CGA<75> mk221(){ CGA<75> a{};
  a.v[(0*5+0)*5+1]=-GG; a.v[(0*5+1)*5+0]= GG; a.v[(0*5+2)*5+3]= G3;
  a.v[(0*5+3)*5+2]=-G3; a.v[(0*5+3)*5+4]=-GG; a.v[(0*5+4)*5+3]= GG;
  a.v[(1*5+0)*5+4]= 2*GG; a.v[(1*5+1)*5+3]=-GG; a.v[(1*5+3)*5+1]= GG; a.v[(1*5+4)*5+0]=-2*GG;
  a.v[(2*5+0)*5+3]= GG; a.v[(2*5+1)*5+2]= G3; a.v[(2*5+1)*5+4]=-GG;
  a.v[(2*5+2)*5+1]=-G3; a.v[(2*5+3)*5+0]=-GG; a.v[(2*5+4)*5+1]= GG; return a; }
// (2,2,2): symmetric coupling pattern, norm 1; idx (k*5+K)*5+Kp
constexpr CGA<125> mk222(){ CGA<125> a{};
  a.v[(0*5+0)*5+2]= HH; a.v[(0*5+2)*5+0]= HH;
  a.v[(1*5+1)*5+2]= HH; a.v[(1*5+2)*5+1]= HH;
  a.v[(3*5+3)*5+2]= HH; a.v[(3*5+2)*5+3]= HH;
  a.v[(4*5+4)*5+2]= HH; a.v[(4*5+2)*5+4]= HH;
  a.v[(2*5+2)*5+2]=-HH;
  a.v[(0*5+1)*5+3]= HH; a.v[(0*5+3)*5+1]= HH;
  a.v[(1*5+0)*5+3]= HH; a.v[(1*5+3)*5+0]= HH;
  a.v[(3*5+0)*5+1]= HH; a.v[(3*5+1)*5+0]= HH; return a; }

__constant__ CGA<1>   CG000 = mk000();
__constant__ CGA<9>   CG011 = mk011();
__constant__ CGA<25>  CG022 = mk022();
__constant__ CGA<9>   CG101 = mk101();
__constant__ CGA<9>   CG110 = mk110();
__constant__ CGA<25>  CG202 = mk202();
__constant__ CGA<25>  CG220 = mk220();
__constant__ CGA<27>  CG111 = mk111();
__constant__ CGA<45>  CG112 = mk112();
__constant__ CGA<45>  CG211 = mk211();
__constant__ CGA<75>  CG212 = mk212();
__constant__ CGA<75>  CG221 = mk221();
__constant__ CGA<125> CG222 = mk222();

// ------------------------- generic WMMA f32 GEMM -------------------------
// C[M,N] = act(scale * A[M,K] @ B[K,N]); one wave (32 thr) per 16x16 tile.
// Edge tiles: A-row / B-col indices are clamped; garbage values only land in
// C rows/cols that are never stored (WMMA rows/cols don't mix), so the K loop
// is completely branchless. K must be a multiple of 8 (true for all calls).
// A fragment (16x4 f32): lanes 0-15 row M=lane, K=k0,k0+1; lanes 16-31 K=k0+2,k0+3.
// B fragment (4x16): vgpr j: lanes 0-15 -> B[k0+j][col], lanes 16-31 -> B[k0+2+j][col].
// C/D: vgpr r: lanes 0-15 -> row r, lanes 16-31 -> row r+8 (col = lane&15).
__global__ __launch_bounds__(32) void wmma_gemm_f32(
    const float* __restrict__ A, const float* __restrict__ B, float* __restrict__ C,
    int M, int K, int N, float scale, int act)
{
  const int tile_m = blockIdx.x * 16;
  const int tile_n = blockIdx.y * 16;
  const int lane = threadIdx.x;
  const int half = lane >> 4;
  const int l16  = lane & 15;
  const int arow_u = tile_m + l16;
  const int bcol_u = tile_n + l16;
  const int arow = clampi(arow_u, M - 1);
  const int bcol = clampi(bcol_u, N - 1);
  const float* __restrict__ Arow = A + (size_t)arow * K;
  const float* __restrict__ Bp   = B + bcol;
  v8f acc = {};
  for (int k0 = 0; k0 < K; k0 += 8) {
    {
      const int ka = k0 + half * 2;
      v2f a = *(const v2f*)(Arow + ka);                 // one b64 load
      v2f b;
      b.x = Bp[(size_t)ka * N];
      b.y = Bp[(size_t)(ka + 1) * N];
      acc = __builtin_amdgcn_wmma_f32_16x16x4_f32(false, a, false, b, (short)0, acc, false, false);
    }
    {
      const int ka = k0 + 4 + half * 2;
      v2f a = *(const v2f*)(Arow + ka);
      v2f b;
      b.x = Bp[(size_t)ka * N];
      b.y = Bp[(size_t)(ka + 1) * N];
      acc = __builtin_amdgcn_wmma_f32_16x16x4_f32(false, a, false, b, (short)0, acc, false, false);
    }
  }
  #pragma unroll
  for (int r = 0; r < 8; ++r) {
    const int row = tile_m + half * 8 + r;
    if (row < M && bcol_u < N) {
      float v = acc[r] * scale;
      if (act == 1) v = silu_f(v);
      C[(size_t)row * N + bcol_u] = v;
    }
  }
}

// ------------------------- fused fctp_scalar WMMA GEMM -------------------------
// out[n,w,i] = scale * sum_{u,v} x[n,u,i] * z[n,v] * W[u,v,w]
// rows = n*dl + i, K = mul_in*8, A[row,k] = x[n, k>>3, i] * z[n, k&7] on the fly.
// Within an 8-wide K block, u = k>>3 is the same for both lane halves and both
// sub-steps, and each lane only ever touches z[2h], z[2h+1], z[4+2h], z[4+2h+1]
// (h = lane>>4): one x load + 2 WMMA per 8 K, 4 z loads total per kernel.
__global__ __launch_bounds__(32) void fctp_wmma(
    const float* __restrict__ x, const float* __restrict__ z, const float* __restrict__ W,
    float* __restrict__ out, int nnodes, int mul_in, int mul_out, int dl, float scale)
{
  const int M = nnodes * dl;
  const int K = mul_in * 8;
  const int N = mul_out;
  const int tile_m = blockIdx.x * 16;
  const int tile_n = blockIdx.y * 16;
  const int lane = threadIdx.x;
  const int half = lane >> 4;
  const int l16  = lane & 15;
  const int arow = clampi(tile_m + l16, M - 1);
  const int n = arow / dl;
  const int i = arow - n * dl;
  const float* __restrict__ xrow = x + (size_t)n * mul_in * dl + i;
  const float* __restrict__ zn = z + (size_t)n * 8;
  const float za0 = zn[half * 2],     za1 = zn[half * 2 + 1];
  const float zb0 = zn[4 + half * 2], zb1 = zn[4 + half * 2 + 1];
  const int bcol_u = tile_n + l16;
  const int bcol = clampi(bcol_u, N - 1);
  const float* __restrict__ Bp = W + bcol;
  v8f acc = {};
  for (int k0 = 0; k0 < K; k0 += 8) {
    const float xv = xrow[(size_t)(k0 >> 3) * dl];      // one x load per 8 K
    {
      const int ka = k0 + half * 2;
      v2f a, b;
      a.x = xv * za0; a.y = xv * za1;
      b.x = Bp[(size_t)ka * N];
      b.y = Bp[(size_t)(ka + 1) * N];
      acc = __builtin_amdgcn_wmma_f32_16x16x4_f32(false, a, false, b, (short)0, acc, false, false);
    }
    {
      const int ka = k0 + 4 + half * 2;
      v2f a, b;
      a.x = xv * zb0; a.y = xv * zb1;
      b.x = Bp[(size_t)ka * N];
      b.y = Bp[(size_t)(ka + 1) * N];
      acc = __builtin_amdgcn_wmma_f32_16x16x4_f32(false, a, false, b, (short)0, acc, false, false);
    }
  }
  // incremental (nn, ii) instead of per-r division
  int row = tile_m + half * 8;
  int nn = row / dl;
  int ii = row - nn * dl;
  #pragma unroll
  for (int r = 0; r < 8; ++r) {
    if (row < M && bcol_u < N)
      out[((size_t)nn * mul_out + bcol_u) * dl + ii] = acc[r] * scale;
    ++row; ++ii;
    if (ii == dl) { ii = 0; ++nn; }
  }
}

// ------------------------- time embedding (single block) -------------------------
__global__ void time_embed_kernel(const float* __restrict__ t,
    const float* __restrict__ w1, const float* __restrict__ b1,
    const float* __restrict__ w2, const float* __restrict__ b2,
    const float* __restrict__ pw, const float* __restrict__ pb,
    float* __restrict__ ht)
{
  __shared__ float basis[12];
  __shared__ float h1[128];
  __shared__ float h2[64];
  const int tid = threadIdx.x;   // 128 threads
  const float tv = t[0];
  if (tid < 12) { float m = (float)tid / 11.0f; float d = (tv - m) / 0.1f; basis[tid] = __expf(-0.5f * d * d); }
  __syncthreads();
  { float a = b1[tid];
    for (int j = 0; j < 12; ++j) a += basis[j] * w1[j * 128 + tid];
    h1[tid] = softplus_f(a); }
  __syncthreads();
  if (tid < 64) { float a = b2[tid];
    for (int j = 0; j < 128; ++j) a += h1[j] * w2[j * 64 + tid];
    h2[tid] = a; }
  __syncthreads();
  for (int c = tid; c < 320; c += 128) {
    float a = pb[c];
    for (int j = 0; j < 64; ++j) a += h2[j] * pw[j * 320 + c];
    ht[c] = a;
  }
}

// ------------------------- tensor product + scatter -------------------------
__device__ __forceinline__ void sph_harm_dev(const float* ev, float* sh1, float* sh2)
{
  const float vx = ev[0], vy = ev[1], vz = ev[2];
  const float r = sqrtf(vx * vx + vy * vy + vz * vz) + 1e-12f;
  const float x = vx / r, y = vy / r, z = vz / r;
  const float S15 = 3.8729833f, S5 = 2.2360680f, S3 = 1.7320508f;
  sh1[0] = S3 * y; sh1[1] = S3 * z; sh1[2] = S3 * x;
  sh2[0] = S15 * x * y; sh2[1] = S15 * y * z; sh2[2] = 0.5f * S5 * (3.f * z * z - 1.f);
  sh2[3] = S15 * x * z; sh2[4] = 0.5f * S15 * (x * x - y * y);
}

// layer 1: PATHS1 = [(0,0,0,8),(0,1,1,8),(0,2,2,8)], xe stride 8, w numel 24
__global__ void tp1_kernel(const float* __restrict__ y0, const float* __restrict__ wbuf,
    const float* __restrict__ evec, const int* __restrict__ src, const int* __restrict__ dst,
    float* __restrict__ agg0, float* __restrict__ agg1, float* __restrict__ agg2, int E)
{
  const int e = blockIdx.x * blockDim.x + threadIdx.x;
  if (e >= E) return;
  float sh1[3], sh2[5];
  sph_harm_dev(evec + 3 * (size_t)e, sh1, sh2);
  const int s = src[e], d = dst[e];
  const float* w = wbuf + (size_t)e * 24;
  const float inv12 = 0.28867513f;  // (cg-norm * sqrt(2l3+1)) cancels; /sqrt(12) neighbors
  for (int u = 0; u < 8; ++u) {
    const float xv = y0[(size_t)s * 8 + u];
    atomicAdd(&agg0[(size_t)d * 8 + u], xv * w[u] * inv12);
    const float w1v = xv * w[8 + u] * inv12;
    for (int k = 0; k < 3; ++k) atomicAdd(&agg1[((size_t)d * 8 + u) * 3 + k], w1v * sh1[k]);
    const float w2v = xv * w[16 + u] * inv12;
    for (int k = 0; k < 5; ++k) atomicAdd(&agg2[((size_t)d * 8 + u) * 5 + k], w2v * sh2[k]);
  }
}

template<int DL1, int DL2, int DL3>
__device__ __forceinline__ void tp2_path(const float* cg, const float* __restrict__ xe,
    const float* sh, const float* __restrict__ w, int mul, float pscale,
    float* __restrict__ agg, int CH, int chan_off, int d)
{
  float m[DL3 * DL1];
  #pragma unroll
  for (int k = 0; k < DL3; ++k)
    #pragma unroll
    for (int i = 0; i < DL1; ++i) {
      float a = 0.f;
      #pragma unroll
      for (int j = 0; j < DL2; ++j) a += cg[(k * DL1 + i) * DL2 + j] * sh[j];
      m[k * DL1 + i] = a;
    }
  for (int u = 0; u < mul; ++u) {
    const float wv = w[u] * pscale;
    #pragma unroll
    for (int k = 0; k < DL3; ++k) {
      float acc = 0.f;
      #pragma unroll
      for (int i = 0; i < DL1; ++i) acc += xe[u * DL1 + i] * m[k * DL1 + i];
      atomicAdd(&agg[((size_t)d * CH + chan_off + u) * DL3 + k], acc * wv);
    }
  }
}

// layers 2/3: PATHS2, xe strides 64/96/160, w numel 512
__global__ void tp2_kernel(const float* __restrict__ y0, const float* __restrict__ y1,
    const float* __restrict__ y2, const float* __restrict__ wbuf,
    const float* __restrict__ evec, const int* __restrict__ src, const int* __restrict__ dst,
    float* __restrict__ agg0, float* __restrict__ agg1, float* __restrict__ agg2, int E)
{
  const int e = blockIdx.x * blockDim.x + threadIdx.x;
  if (e >= E) return;
  float sh0[1] = {1.0f};
  float sh1[3], sh2[5];
  sph_harm_dev(evec + 3 * (size_t)e, sh1, sh2);
  const int s = src[e], d = dst[e];
  const float* w = wbuf + (size_t)e * 512;
  const float* xe0 = y0 + (size_t)s * 64;
  const float* xe1 = y1 + (size_t)s * 96;
  const float* xe2 = y2 + (size_t)s * 160;
  const float P0 = 0.16666667f;  // sqrt(1)/(sqrt(3)*sqrt(12))
  const float P1 = 0.22360680f;  // sqrt(3)/(sqrt(5)*sqrt(12))
  const float P2 = 0.28867513f;  // sqrt(5)/(sqrt(5)*sqrt(12))
  tp2_path<1,1,1>(CG000.v, xe0, sh0, w +   0, 64, P0, agg0, 128,   0, d);
  tp2_path<1,3,3>(CG011.v, xe0, sh1, w +  64, 64, P1, agg1, 192,   0, d);
  tp2_path<1,5,5>(CG022.v, xe0, sh2, w + 128, 64, P2, agg2, 192,   0, d);
  tp2_path<3,1,3>(CG101.v, xe1, sh0, w + 192, 32, P1, agg1, 192,  64, d);
  tp2_path<3,3,1>(CG110.v, xe1, sh1, w + 224, 32, P0, agg0, 128,  64, d);
  tp2_path<3,3,3>(CG111.v, xe1, sh1, w + 256, 32, P1, agg1, 192,  96, d);
  tp2_path<3,3,5>(CG112.v, xe1, sh1, w + 288, 32, P2, agg2, 192,  64, d);
  tp2_path<5,1,5>(CG202.v, xe2, sh0, w + 320, 32, P2, agg2, 192,  96, d);
  tp2_path<5,3,3>(CG211.v, xe2, sh1, w + 352, 32, P1, agg1, 192, 128, d);
  tp2_path<5,3,5>(CG212.v, xe2, sh1, w + 384, 32, P2, agg2, 192, 128, d);
  tp2_path<5,5,1>(CG220.v, xe2, sh2, w + 416, 32, P0, agg0, 128,  96, d);
  tp2_path<5,5,3>(CG221.v, xe2, sh2, w + 448, 32, P1, agg1, 192, 160, d);
  tp2_path<5,5,5>(CG222.v, xe2, sh2, w + 480, 32, P2, agg2, 192, 160, d);
}

// ------------------------- combine + gate + time add -------------------------
__global__ void gate_kernel(const float* __restrict__ sc0, const float* __restrict__ sc1,
    const float* __restrict__ sc2, const float* __restrict__ x20, const float* __restrict__ x21,
    const float* __restrict__ x22, const float* __restrict__ ht,
    float* __restrict__ X0, float* __restrict__ X1, float* __restrict__ X2,
    int N, int has_sc12)
{
  const int n = blockIdx.x * blockDim.x + threadIdx.x;
  if (n >= N) return;
  const float cs = 0.38268343f, cx = 0.92387953f;  // sin/cos(pi/8)
  const float* s0 = sc0 + (size_t)n * 128;
  const float* a0 = x20 + (size_t)n * 128;
  for (int c = 0; c < 64; ++c) {
    const float v = cs * s0[c] + cx * a0[c];
    X0[(size_t)n * 64 + c] = silu_f(v) + ht[c];
  }
  for (int u = 0; u < 32; ++u) {
    const float g1 = sigmoid_f(cs * s0[64 + u] + cx * a0[64 + u]);
    const float g2 = sigmoid_f(cs * s0[96 + u] + cx * a0[96 + u]);
    for (int k = 0; k < 3; ++k) {
      float v = x21[(size_t)n * 96 + u * 3 + k];
      if (has_sc12) v = cs * sc1[(size_t)n * 96 + u * 3 + k] + cx * v;
      X1[(size_t)n * 96 + u * 3 + k] = v * g1 + ht[64 + u * 3 + k];
    }
    for (int k = 0; k < 5; ++k) {
      float v = x22[(size_t)n * 160 + u * 5 + k];
      if (has_sc12) v = cs * sc2[(size_t)n * 160 + u * 5 + k] + cx * v;
      X2[(size_t)n * 160 + u * 5 + k] = v * g2 + ht[160 + u * 5 + k];
    }
  }
}

// ------------------------- final projection -------------------------
__global__ void out_kernel(const float* __restrict__ X1, const float* __restrict__ z,
    const float* __restrict__ wout, float* __restrict__ out, int N)
{
  const int n = blockIdx.x * blockDim.x + threadIdx.x;
  if (n >= N) return;
  float zr[8];
  #pragma unroll
  for (int v = 0; v < 8; ++v) zr[v] = z[n * 8 + v];
  float o[3] = {0.f, 0.f, 0.f};
  for (int u = 0; u < 32; ++u) {
    float wu = 0.f;
    #pragma unroll
    for (int v = 0; v < 8; ++v) wu += zr[v] * wout[u * 8 + v];
    for (int k = 0; k < 3; ++k) o[k] += X1[(size_t)n * 96 + u * 3 + k] * wu;
  }
  for (int k = 0; k < 3; ++k) out[(size_t)n * 3 + k] = o[k] * 0.0625f;  // 1/sqrt(32*8)
}

// ---------------------------------------------------------------------------
extern "C" void kernel_launch(void* const* d_in, const int* in_sizes, int n_in,
                              void* d_out, int out_size, void* d_ws, size_t ws_size,
                              hipStream_t stream)
{
  // Assumed flatten order: jax pytree (sorted dict keys):
  // 0 h_node_x, 1 h_node_z, 2 h_edge, 3 edge_vec, 4 t, 5 edge_src, 6 edge_dst,
  // params1: 7 lin1[0], 8 lin2[0], 9 lin2[1], 10 lin2[2], 11 rad_w1, 12 rad_w2, 13 sc[0]
  // params2: 14 lin1[0..2], 17 lin2[0..2], 20 rad_w1, 21 rad_w2, 22 sc[0..2]
  // params3: 25..35 (same layout), 36 w_out,
  // tparams: 37 b1, 38 b2, 39 pb, 40 pw, 41 w1, 42 w2
  const float* h_node_x = (const float*)d_in[0];
  const float* z        = (const float*)d_in[1];
  const float* h_edge   = (const float*)d_in[2];
  const float* edge_vec = (const float*)d_in[3];
  const float* t_in     = (const float*)d_in[4];
  const int*   esrc     = (const int*)d_in[5];
  const int*   edst     = (const int*)d_in[6];
  const int N = in_sizes[0] / 8;
  const int E = in_sizes[5];

  struct Lay { const float *l10,*l11,*l12,*l20,*l21,*l22,*rw1,*rw2,*sc0,*sc1,*sc2; };
  Lay L1 = { (const float*)d_in[7], nullptr, nullptr,
             (const float*)d_in[8], (const float*)d_in[9], (const float*)d_in[10],
             (const float*)d_in[11], (const float*)d_in[12],
             (const float*)d_in[13], nullptr, nullptr };
  Lay L2 = { (const float*)d_in[14], (const float*)d_in[15], (const float*)d_in[16],
             (const float*)d_in[17], (const float*)d_in[18], (const float*)d_in[19],
             (const float*)d_in[20], (const float*)d_in[21],
             (const float*)d_in[22], (const float*)d_in[23], (const float*)d_in[24] };
  Lay L3 = { (const float*)d_in[25], (const float*)d_in[26], (const float*)d_in[27],
             (const float*)d_in[28], (const float*)d_in[29], (const float*)d_in[30],
             (const float*)d_in[31], (const float*)d_in[32],
             (const float*)d_in[33], (const float*)d_in[34], (const float*)d_in[35] };
  const float* wout = (const float*)d_in[36];
  const float* tb1  = (const float*)d_in[37];
  const float* tb2  = (const float*)d_in[38];
  const float* tpb  = (const float*)d_in[39];
  const float* tpw  = (const float*)d_in[40];
  const float* tw1  = (const float*)d_in[41];
  const float* tw2  = (const float*)d_in[42];

  // ---------------- workspace layout (floats) ----------------
  float* ws = (float*)d_ws;
  size_t off = 0;
  auto alloc = [&](size_t n) { float* p = ws + off; off += n; return p; };
  float* HT   = alloc(320);
  float* X0   = alloc((size_t)N * 64);
  float* X1   = alloc((size_t)N * 96);
  float* X2   = alloc((size_t)N * 160);
  float* SC0  = alloc((size_t)N * 128);
  float* SC1  = alloc((size_t)N * 96);
  float* SC2  = alloc((size_t)N * 160);
  float* Y0   = alloc((size_t)N * 64);
  float* Y1   = alloc((size_t)N * 96);
  float* Y2   = alloc((size_t)N * 160);
  float* AG0  = alloc((size_t)N * 128);
  float* AG1  = alloc((size_t)N * 576);
  float* AG2  = alloc((size_t)N * 960);
  float* X20  = alloc((size_t)N * 128);
  float* X21  = alloc((size_t)N * 96);
  float* X22  = alloc((size_t)N * 160);
  float* HID  = alloc((size_t)E * 64);
  float* WBUF = alloc((size_t)E * 512);
  (void)ws_size; (void)n_in; (void)out_size;

  auto g16 = [](int M, int Nc) { return dim3((unsigned)((M + 15) / 16), (unsigned)((Nc + 15) / 16), 1); };
  const dim3 blk32(32, 1, 1);
  const int ME = E, M0 = N, M1 = N * 3, M2 = N * 5;
  const float SC8 = 0.125f;          // 1/sqrt(8*8)
  const float SC64 = 0.04419417f;    // 1/sqrt(64*8)
  const float SC32 = 0.0625f;        // 1/sqrt(32*8)
  const float SC128 = 0.03125f;      // 1/sqrt(128*8)
  const float SC192 = 0.02551552f;   // 1/sqrt(192*8)

  // time embedding (t is a scalar; ht is a single 320-vector broadcast to all nodes)
  time_embed_kernel<<<1, 128, 0, stream>>>(t_in, tw1, tb1, tw2, tb2, tpw, tpb, HT);

  // ============ layer 1 (PATHS1, scalar-only input x = h_node_x[:, :8]) ============
  wmma_gemm_f32<<<g16(ME, 64), blk32, 0, stream>>>(h_edge, L1.rw1, HID, ME, 16, 64, 1.0f, 1);
  wmma_gemm_f32<<<g16(ME, 24), blk32, 0, stream>>>(HID, L1.rw2, WBUF, ME, 64, 24, 1.0f, 0);
  fctp_wmma<<<g16(M0, 128), blk32, 0, stream>>>(h_node_x, z, L1.sc0, SC0, N, 8, 128, 1, SC8);
  fctp_wmma<<<g16(M0,   8), blk32, 0, stream>>>(h_node_x, z, L1.l10, Y0, N, 8, 8, 1, SC8);
  hipMemsetAsync(AG0, 0, (size_t)N * 128 * sizeof(float), stream);
  hipMemsetAsync(AG1, 0, (size_t)N * 576 * sizeof(float), stream);
  hipMemsetAsync(AG2, 0, (size_t)N * 960 * sizeof(float), stream);
  tp1_kernel<<<(E + 127) / 128, 128, 0, stream>>>(Y0, WBUF, edge_vec, esrc, edst, AG0, AG1, AG2, E);
  fctp_wmma<<<g16(M0, 128), blk32, 0, stream>>>(AG0, z, L1.l20, X20, N, 8, 128, 1, SC8);
  fctp_wmma<<<g16(M1,  32), blk32, 0, stream>>>(AG1, z, L1.l21, X21, N, 8, 32, 3, SC8);
  fctp_wmma<<<g16(M2,  32), blk32, 0, stream>>>(AG2, z, L1.l22, X22, N, 8, 32, 5, SC8);
  gate_kernel<<<(N + 127) / 128, 128, 0, stream>>>(SC0, SC1, SC2, X20, X21, X22, HT, X0, X1, X2, N, 0);

  // ============ layers 2 & 3 (PATHS2) ============
  const Lay* Ls[2] = { &L2, &L3 };
  for (int li = 0; li < 2; ++li) {
    const Lay& P = *Ls[li];
    wmma_gemm_f32<<<g16(ME, 64), blk32, 0, stream>>>(h_edge, P.rw1, HID, ME, 16, 64, 1.0f, 1);
    wmma_gemm_f32<<<g16(ME, 512), blk32, 0, stream>>>(HID, P.rw2, WBUF, ME, 64, 512, 1.0f, 0);
    fctp_wmma<<<g16(M0, 128), blk32, 0, stream>>>(X0, z, P.sc0, SC0, N, 64, 128, 1, SC64);
    fctp_wmma<<<g16(M1,  32), blk32, 0, stream>>>(X1, z, P.sc1, SC1, N, 32, 32, 3, SC32);
    fctp_wmma<<<g16(M2,  32), blk32, 0, stream>>>(X2, z, P.sc2, SC2, N, 32, 32, 5, SC32);
    fctp_wmma<<<g16(M0,  64), blk32, 0, stream>>>(X0, z, P.l10, Y0, N, 64, 64, 1, SC64);
    fctp_wmma<<<g16(M1,  32), blk32, 0, stream>>>(X1, z, P.l11, Y1, N, 32, 32, 3, SC32);
    fctp_wmma<<<g16(M2,  32), blk32, 0, stream>>>(X2, z, P.l12, Y2, N, 32, 32, 5, SC32);
    hipMemsetAsync(AG0, 0, (size_t)N * 128 * sizeof(float), stream);
    hipMemsetAsync(AG1, 0, (size_t)N * 576 * sizeof(float), stream);
    hipMemsetAsync(AG2, 0, (size_t)N * 960 * sizeof(float), stream);
    tp2_kernel<<<(E + 127) / 128, 128, 0, stream>>>(Y0, Y1, Y2, WBUF, edge_vec, esrc, edst, AG0, AG1, AG2, E);
    fctp_wmma<<<g16(M0, 128), blk32, 0, stream>>>(AG0, z, P.l20, X20, N, 128, 128, 1, SC128);
    fctp_wmma<<<g16(M1,  32), blk32, 0, stream>>>(AG1, z, P.l21, X21, N, 192, 32, 3, SC192);
    fctp_wmma<<<g16(M2,  32), blk32, 0, stream>>>(AG2, z, P.l22, X22, N, 192, 32, 5, SC192);
    gate_kernel<<<(N + 127) / 128, 128, 0, stream>>>(SC0, SC1, SC2, X20, X21, X22, HT, X0, X1, X2, N, 1);
  }

  // ============ output projection ============
  out_kernel<<<(N + 127) / 128, 128, 0, stream>>>(X1, z, wout, (float*)d_out, N);
}